// Router_26637387169870
// MI455X (gfx1250) — compile-verified
//
#include <hip/hip_runtime.h>
#include <math.h>

typedef __attribute__((ext_vector_type(2))) float v2f;
typedef __attribute__((ext_vector_type(8))) float v8f;

union V8 { v8f v; float f[8]; };

// ---------------------------------------------------------------------------
// Kernel 1: logits[T,8] = x[T,D] @ wg[8,D]^T  +  gr[T,8] @ gm[8,8]^T
// One wave handles 32 tokens (two 16-row WMMA tiles sharing B loads).
// fp32 WMMA 16x16x4. Experts occupy N columns 0-7; columns 8-15 MIRROR
// experts 0-7 (lane n loads wg[n&7]) so every load is unconditional --
// duplicated columns are simply never stored. No exec-mask predication
// in the hot loop.
// ---------------------------------------------------------------------------
__global__ __launch_bounds__(128) void router_gemm_wmma(
    const float* __restrict__ x,
    const float* __restrict__ wg,
    const float* __restrict__ gr,
    const float* __restrict__ gm,
    float* __restrict__ logits,
    int D)
{
  const int lane  = (int)(threadIdx.x & 31u);
  const int wave  = (int)(blockIdx.x * (blockDim.x >> 5) + (threadIdx.x >> 5));
  const int n     = lane & 15;          // N column for B/C/D
  const int e     = n & 7;              // expert actually loaded (mirrored)
  const int khalf = (lane >> 4) << 1;   // K offset: 0 (lanes 0-15) / 2 (lanes 16-31)
  const int tile  = wave * 32;          // first token of this wave

  const size_t r0 = (size_t)(tile + n); // A row for tile 0 (M = lane&15)
  const size_t r1 = r0 + 16;            // A row for tile 1

  const float* xp0 = x  + r0 * (size_t)D + khalf;
  const float* xp1 = x  + r1 * (size_t)D + khalf;
  const float* bp  = wg + (size_t)e * (size_t)D + khalf;

  v8f acc0 = {};
  v8f acc1 = {};

  // Main router GEMM over D, chunked for prefetch + unroll.
  for (int k = 0; k < D; k += 32) {
    if (k + 1056 <= D) {
      __builtin_prefetch(xp0 + k + 1024, 0, 1);
      __builtin_prefetch(xp1 + k + 1024, 0, 1);
    }
#pragma unroll
    for (int kk = 0; kk < 32; kk += 4) {
      v2f a0 = *(const v2f*)(xp0 + k + kk);
      v2f a1 = *(const v2f*)(xp1 + k + kk);
      v2f b  = *(const v2f*)(bp  + k + kk);
      acc0 = __builtin_amdgcn_wmma_f32_16x16x4_f32(false, a0, false, b,
                                                   (short)0, acc0, false, false);
      acc1 = __builtin_amdgcn_wmma_f32_16x16x4_f32(false, a1, false, b,
                                                   (short)0, acc1, false, false);
    }
  }

  // gate_residual @ gate_map^T : K=8 -> two more WMMA steps, same accumulators.
  // B[k=ke][n=f] = gm[f][ke]  (gm is [E,E] row-major, stride 8; mirrored cols)
  {
    const float* ap0 = gr + r0 * 8 + khalf;
    const float* ap1 = gr + r1 * 8 + khalf;
    const float* gp  = gm + (size_t)e * 8 + khalf;
#pragma unroll
    for (int k = 0; k < 8; k += 4) {
      v2f a0 = *(const v2f*)(ap0 + k);
      v2f a1 = *(const v2f*)(ap1 + k);
      v2f b  = *(const v2f*)(gp + k);
      acc0 = __builtin_amdgcn_wmma_f32_16x16x4_f32(false, a0, false, b,
                                                   (short)0, acc0, false, false);
      acc1 = __builtin_amdgcn_wmma_f32_16x16x4_f32(false, a1, false, b,
                                                   (short)0, acc1, false, false);
    }
  }

  // Store C: lanes 0-15 VGPR v -> M=v; lanes 16-31 VGPR v -> M=v+8; col = lane&15.
  // Only the real expert columns (n < 8) are written.
  if (n < 8) {
    V8 c0; c0.v = acc0;
    V8 c1; c1.v = acc1;
    const int mbase = tile + ((lane >> 4) << 3);
#pragma unroll
    for (int v = 0; v < 8; ++v) {
      logits[(size_t)(mbase + v)      * 8 + n] = c0.f[v];
      logits[(size_t)(mbase + 16 + v) * 8 + n] = c1.f[v];
    }
  }
}

// ---------------------------------------------------------------------------
// Kernel 2: per-token gating. std(ddof=1) scale, softmax, stable top-2,
// zero expert E-1, renormalize. One thread per token.
// ---------------------------------------------------------------------------
__global__ __launch_bounds__(256) void router_gate_epilogue(
    const float* __restrict__ logits,
    float* __restrict__ gates,
    float* __restrict__ indices,
    int T)
{
  int t = blockIdx.x * blockDim.x + threadIdx.x;
  if (t >= T) return;

  const float4* lp = (const float4*)(logits + (size_t)t * 8);
  float4 lo = lp[0], hi = lp[1];
  float l[8] = {lo.x, lo.y, lo.z, lo.w, hi.x, hi.y, hi.z, hi.w};

  float s = 0.f;
#pragma unroll
  for (int e = 0; e < 8; ++e) s += l[e];
  float mean = s * 0.125f;

  float var = 0.f;
#pragma unroll
  for (int e = 0; e < 8; ++e) { float d = l[e] - mean; var += d * d; }
  var *= (1.0f / 7.0f);                 // unbiased (ddof=1)
  float inv_std = 1.0f / sqrtf(var);    // MOE_GATE_NORM_STD = 1.0

  float z[8];
  float m = -INFINITY;
#pragma unroll
  for (int e = 0; e < 8; ++e) { z[e] = l[e] * inv_std; m = fmaxf(m, z[e]); }

  float se = 0.f;
#pragma unroll
  for (int e = 0; e < 8; ++e) { z[e] = expf(z[e] - m); se += z[e]; }
  float rse = 1.0f / se;

  // Stable top-2 (descending, earliest index wins ties), like jax.lax.top_k.
  float b0 = -INFINITY, b1 = -INFINITY;
  int   i0 = 0,         i1 = 0;
#pragma unroll
  for (int e = 0; e < 8; ++e) {
    float p = z[e] * rse;
    if (p > b0)      { b1 = b0; i1 = i0; b0 = p; i0 = e; }
    else if (p > b1) { b1 = p;  i1 = e; }
  }

  float g0 = (i0 == 7) ? 0.f : b0;
  float g1 = (i1 == 7) ? 0.f : b1;
  float rg = 1.0f / (g0 + g1);

  gates[(size_t)t * 2 + 0]   = g0 * rg;
  gates[(size_t)t * 2 + 1]   = g1 * rg;
  indices[(size_t)t * 2 + 0] = (float)i0;
  indices[(size_t)t * 2 + 1] = (float)i1;
}

// ---------------------------------------------------------------------------
extern "C" void kernel_launch(void* const* d_in, const int* in_sizes, int n_in,
                              void* d_out, int out_size, void* d_ws, size_t ws_size,
                              hipStream_t stream) {
  const float* x  = (const float*)d_in[0];   // [T, D]
  const float* wg = (const float*)d_in[1];   // [E, D]
  const float* gm = (const float*)d_in[2];   // [E, E]
  const float* gr = (const float*)d_in[3];   // [T, E]

  const int E = 8;
  const int T = in_sizes[3] / E;             // 16384
  const int D = in_sizes[0] / T;             // 2048

  // d_out layout (tuple concatenated): gates [T,2] | indices [T,2] | logits [T,8]
  float* gates   = (float*)d_out;
  float* indices = gates + (size_t)T * 2;
  float* logits  = indices + (size_t)T * 2;

  const int wavesPerBlock = 4;               // 128 threads
  const int waves = T / 32;                  // 512 waves, 32 tokens each
  router_gemm_wmma<<<dim3(waves / wavesPerBlock), dim3(32 * wavesPerBlock),
                     0, stream>>>(x, wg, gr, gm, logits, D);

  router_gate_epilogue<<<dim3((T + 255) / 256), dim3(256), 0, stream>>>(
      logits, gates, indices, T);
}